// MixtureCDFFlow_62285615727067
// MI455X (gfx1250) — compile-verified
//
#include <hip/hip_runtime.h>
#include <math.h>

// Problem constants from the reference
#define KMIX 8
#define DDIM 16
#define NPARAM (KMIX + 2 * KMIX * DDIM)   // 8 logits + 128 mu + 128 logstd = 264 dwords

#define SQRT1_2F   0.70710678118654752f   // 1/sqrt(2)
#define INV_S2PI_F 0.39894228040143268f   // 1/sqrt(2*pi)

// LDS slot layout: [0,8): logits, [8,136): mu[K][D], [136,264): logstd[K][D]
#define SLOT_MU     KMIX
#define SLOT_LOGSTD (KMIX + KMIX * DDIM)

__global__ __launch_bounds__(256)
void mixflow_invert_kernel(const float* __restrict__ z,
                           const float* __restrict__ logits,
                           const float* __restrict__ mu,
                           const float* __restrict__ logstd,
                           float* __restrict__ out_x,
                           float* __restrict__ out_nld,
                           int n)
{
    // Sole shared allocation -> group-segment offset 0, so the LDS byte
    // address of sp[s] is simply 4*s. Staged once per workgroup with CDNA5
    // async global->LDS loads (tracked by ASYNCcnt).
    __shared__ float sp[NPARAM];

    const int t   = threadIdx.x;
    const int idx = (int)blockIdx.x * (int)blockDim.x + t;

    // ---- CDNA5 async global -> LDS staging (uniform, three issues) ---------
    if (t < KMIX) {
        const float* gp = logits + t;
        unsigned lds_addr = 4u * (unsigned)t;
        asm volatile("global_load_async_to_lds_b32 %0, %1, off"
                     :: "v"(lds_addr), "v"(gp) : "memory");
    }
    if (t < KMIX * DDIM) {
        const float* gp = mu + t;
        unsigned lds_addr = 4u * (unsigned)(SLOT_MU + t);
        asm volatile("global_load_async_to_lds_b32 %0, %1, off"
                     :: "v"(lds_addr), "v"(gp) : "memory");
        const float* gp2 = logstd + t;
        unsigned lds_addr2 = 4u * (unsigned)(SLOT_LOGSTD + t);
        asm volatile("global_load_async_to_lds_b32 %0, %1, off"
                     :: "v"(lds_addr2), "v"(gp2) : "memory");
    }

    // Independent global load overlaps with the async staging.
    float zc = (idx < n) ? z[idx] : 0.0f;

    // Wait for our async LDS writes, then make them visible workgroup-wide.
    asm volatile("s_wait_asynccnt 0x0" ::: "memory");
    __syncthreads();

    if (idx >= n) return;

    const int d = idx & (DDIM - 1);
    zc = fminf(fmaxf(zc, 0.0f), 1.0f);

    // ---- per-thread parameter prep (registers) -----------------------------
    float lg[KMIX], muk[KMIX], lsk[KMIX];
#pragma unroll
    for (int k = 0; k < KMIX; ++k) {
        lg[k]  = sp[k];
        muk[k] = sp[SLOT_MU + k * DDIM + d];
        lsk[k] = sp[SLOT_LOGSTD + k * DDIM + d];
    }

    // softmax(logits)
    float mx = lg[0];
#pragma unroll
    for (int k = 1; k < KMIX; ++k) mx = fmaxf(mx, lg[k]);
    float w[KMIX];
    float wsum = 0.0f;
#pragma unroll
    for (int k = 0; k < KMIX; ++k) { w[k] = expf(lg[k] - mx); wsum += w[k]; }
    const float inv_wsum = 1.0f / wsum;

    // derived per-(k,d) constants
    float a[KMIX];    // (1/sigma) * (1/sqrt(2))    -> erf argument scale
    float cw[KMIX];   // 0.5 * w_k                  -> CDF weight
    float p[KMIX];    // w_k * inv_sqrt_2pi / sigma -> pdf coefficient
    float msum = 0.0f, mumin = 3.4e38f, mumax = -3.4e38f;
    float cwsum = 0.0f;
#pragma unroll
    for (int k = 0; k < KMIX; ++k) {
        const float wk    = w[k] * inv_wsum;
        const float scale = expf(lsk[k]);
        const float rcp   = 1.0f / scale;
        a[k]  = rcp * SQRT1_2F;
        cw[k] = 0.5f * wk;
        p[k]  = wk * INV_S2PI_F * rcp;
        msum += scale;
        mumin = fminf(mumin, muk[k]);
        mumax = fmaxf(mumax, muk[k]);
        cwsum += 0.5f * wk;
    }
    float lb = mumin - 10.0f * msum;
    float ub = mumax + 10.0f * msum;

    // ---- bisection (matches reference update; fp32 fixed point <= 200 it) --
    float x = 0.0f;
    for (int it = 0; it < 200; ++it) {
        float cur = cwsum;                 // sum_k 0.5*w_k * (1 + erf(...))
#pragma unroll
        for (int k = 0; k < KMIX; ++k) {
            const float u = (x - muk[k]) * a[k];
            cur = fmaf(cw[k], erff(u), cur);
        }
        const bool gt = (cur > zc);
        const float nx = 0.5f * (x + (gt ? lb : ub));
        if (gt) ub = x; else lb = x;
        if (nx == x) break;                // per-element fp32 fixed point
        x = nx;
    }

    // ---- -log(sum_k w_k * pdf_k(x)) ----------------------------------------
    float s = 0.0f;
#pragma unroll
    for (int k = 0; k < KMIX; ++k) {
        const float u = (x - muk[k]) * a[k];   // = ((x-mu)/sigma)/sqrt(2)
        s = fmaf(p[k], expf(-u * u), s);       // exp(-u^2) = exp(-0.5*((x-mu)/sigma)^2)
    }

    out_x[idx]   = x;
    out_nld[idx] = -logf(s);
}

extern "C" void kernel_launch(void* const* d_in, const int* in_sizes, int n_in,
                              void* d_out, int out_size, void* d_ws, size_t ws_size,
                              hipStream_t stream) {
    const float* z      = (const float*)d_in[0];
    const float* logits = (const float*)d_in[1];
    const float* mu     = (const float*)d_in[2];
    const float* logstd = (const float*)d_in[3];

    const int n = in_sizes[0];              // B*S*D elements
    float* out_x   = (float*)d_out;         // first output: x
    float* out_nld = (float*)d_out + n;     // second output: -log_det

    const int block = 256;
    const int grid  = (n + block - 1) / block;
    mixflow_invert_kernel<<<grid, block, 0, stream>>>(z, logits, mu, logstd,
                                                      out_x, out_nld, n);
}